// RiemannianDiffusionFull_29171417874549
// MI455X (gfx1250) — compile-verified
//
#include <hip/hip_runtime.h>
#include <hip/hip_bf16.h>
#include <math.h>

// Sizes from the reference
#define BATCH 131072
#define DDIM  784
#define HID   256
#define MD    16

#define M_TILE 64           // batch rows per workgroup
#define KC1    56           // K-chunk for GEMM1 (784 = 14*56)
#define KC2    64           // K-chunk for GEMM2 (256 = 4*64)
#define NCH1   (DDIM / KC1) // 14
#define NCH2   (HID / KC2)  // 4
#define XSTR   58           // X tile row stride: 8B-aligned for async B64,
                            // 58*l mod 64 distinct for l<16 -> conflict-free

typedef float v2f __attribute__((ext_vector_type(2)));
typedef float v8f __attribute__((ext_vector_type(8)));

// D = A(16x4,f32) * B(4x16,f32) + C(16x16,f32)
#define WMMA4(a, b, c) \
  __builtin_amdgcn_wmma_f32_16x16x4_f32(false, (a), false, (b), (short)0, (c), false, false)

// LDS float offsets (phase-aliased; total 40192 floats = 160768 B)
#define OFF_XS0  0        // X chunk buf0: 64 x 58 = 3712
#define OFF_XS1  3712     // X chunk buf1: 3712            (-> 7424)
#define OFF_W1S  7424     // W1 chunk: 56 x 256 = 14336    (-> 21760)
#define OFF_WPS  21760    // Wp chunk: 56 x 16  = 896      (-> 22656 = region A end)
#define OFF_HS   0        // phase2+: Hs/gflat 64 x 260 = 16640 (aliases region A)
#define OFF_W2S  22656    // GEMM2: W2 chunk 64 x 256 = 16384
#define OFF_MATS 22656    // solve: Mats 64 x 257 = 16448 (aliases W2s; -> 39104)
#define OFF_SS   39104    // Ss 64 x 17 = 1088 (-> 40192)
#define SMEM_FLOATS 40192

// ---- CDNA5 async memory->LDS copies (ASYNCcnt tracked, in-order completion) ----
__device__ __forceinline__ void async_copy_b128(unsigned lds_byte_addr, const void* gaddr) {
  asm volatile("global_load_async_to_lds_b128 %0, %1, off"
               :: "v"(lds_byte_addr), "v"(gaddr) : "memory");
}
__device__ __forceinline__ void async_copy_b64(unsigned lds_byte_addr, const void* gaddr) {
  asm volatile("global_load_async_to_lds_b64 %0, %1, off"
               :: "v"(lds_byte_addr), "v"(gaddr) : "memory");
}
template <int N>
__device__ __forceinline__ void wait_async_le() {
  asm volatile("s_wait_asynccnt %0" :: "n"(N) : "memory");
}

__global__ __launch_bounds__(256, 1)
void riemannian_diffusion_fused(const float* __restrict__ x,
                                const float* __restrict__ t,
                                const float* __restrict__ W1,
                                const float* __restrict__ b1,
                                const float* __restrict__ W2,
                                const float* __restrict__ b2,
                                const float* __restrict__ Wp,
                                const float* __restrict__ bp,
                                const float* __restrict__ Wl,
                                const float* __restrict__ bl,
                                float* __restrict__ out)
{
  __shared__ float smem[SMEM_FLOATS];
  float* W1s  = smem + OFF_W1S;
  float* Wps  = smem + OFF_WPS;
  float* Hs   = smem + OFF_HS;    // also holds gflat (A matrices) later
  float* W2s  = smem + OFF_W2S;
  float* Mats = smem + OFF_MATS;
  float* Ss   = smem + OFF_SS;

  // LDS byte offset of smem base (generic->LDS: low 32 bits hold the LDS offset)
  const unsigned lds_base = (unsigned)(unsigned long long)(uintptr_t)smem;

  const int tid  = threadIdx.x;
  const int wave = tid >> 5;
  const int lane = tid & 31;
  const int half = lane >> 4;      // 0: lanes 0-15, 1: lanes 16-31
  const int lx   = lane & 15;
  const int rt   = wave & 3;       // row-tile (16 rows each) of the 64-row slab
  const int ch   = wave >> 2;      // column-half
  const int m0   = rt * 16;
  const int row0 = blockIdx.x * M_TILE;

  const v8f vzero = {0.f, 0.f, 0.f, 0.f, 0.f, 0.f, 0.f, 0.f};

  // ---- async stagers --------------------------------------------------------
  // X chunk c into buffer `buf`: 64 rows x 56 floats as B64 => exactly 7 per thread
  auto issue_X = [&](int c, int buf) {
    const unsigned xoff = (buf ? OFF_XS1 : OFF_XS0);
    for (int idx = tid; idx < 64 * 28; idx += 256) {
      const int r  = idx / 28;
      const int j2 = idx % 28;
      async_copy_b64(lds_base + (xoff + (unsigned)(r * XSTR + j2 * 2)) * 4u,
                     x + (size_t)(row0 + r) * DDIM + c * KC1 + j2 * 2);
    }
  };
  // W1 chunk c: 56 x 256 as B128 (14 per thread); Wp chunk c: 56 x 16 (224 B128)
  auto issue_W1Wp = [&](int c) {
    for (int idx = tid; idx < 56 * 64; idx += 256) {
      const int r  = idx >> 6;
      const int c4 = idx & 63;
      async_copy_b128(lds_base + (unsigned)(OFF_W1S + idx * 4) * 4u,
                      W1 + (size_t)(c * KC1 + r) * 256 + c4 * 4);
    }
    for (int idx = tid; idx < 56 * 4; idx += 256) {
      async_copy_b128(lds_base + (unsigned)(OFF_WPS + idx * 4) * 4u,
                      Wp + (size_t)(c * KC1 + (idx >> 2)) * 16 + (idx & 3) * 4);
    }
  };
  auto issue_W2 = [&](int c2) {
    for (int idx = tid; idx < 64 * 64; idx += 256) {
      const int r  = idx >> 6;
      const int c4 = idx & 63;
      async_copy_b128(lds_base + (unsigned)(OFF_W2S + idx * 4) * 4u,
                      W2 + (size_t)(c2 * KC2 + r) * 256 + c4 * 4);
    }
  };

  // ------------------------------------------------------------------
  // Phase 1: H = relu(X @ W1 + b1), XI = X @ Wp + bp (K=784, 14 chunks)
  // X double-buffered: chunk c+1 streams from HBM while chunk c computes.
  // ------------------------------------------------------------------
  v8f acc[8];
#pragma unroll
  for (int i = 0; i < 8; ++i) acc[i] = vzero;
  v8f xiacc = vzero;

  issue_X(0, 0);
  issue_W1Wp(0);
  for (int c = 0; c < NCH1; ++c) {
    if (c + 1 < NCH1) {
      issue_X(c + 1, (c + 1) & 1);   // prefetch next X chunk (newest 7 per thread)
      wait_async_le<7>();            // in-order => chunk c (X+W1+Wp) fully landed
    } else {
      wait_async_le<0>();
    }
    __syncthreads();

    const float* Xsc = smem + (c & 1 ? OFF_XS1 : OFF_XS0);
    for (int k = 0; k < KC1; k += 4) {
      v2f a;
      a.x = Xsc[(m0 + lx) * XSTR + k + half * 2];
      a.y = Xsc[(m0 + lx) * XSTR + k + half * 2 + 1];
#pragma unroll
      for (int nt = 0; nt < 8; ++nt) {
        const int n0 = (ch * 8 + nt) * 16;
        v2f b;
        b.x = W1s[(k + half * 2) * 256 + n0 + lx];
        b.y = W1s[(k + half * 2 + 1) * 256 + n0 + lx];
        acc[nt] = WMMA4(a, b, acc[nt]);
      }
      // xi projection: all waves compute it (no EXEC juggling); only ch==0 writes.
      {
        v2f b;
        b.x = Wps[(k + half * 2) * 16 + lx];
        b.y = Wps[(k + half * 2 + 1) * 16 + lx];
        xiacc = WMMA4(a, b, xiacc);
      }
    }
    __syncthreads();                 // all waves done reading W1s/Wps/this X buf
    if (c + 1 < NCH1) issue_W1Wp(c + 1);
  }

  // bias + ReLU -> Hs (stride 260, overlays staging region); xi -> score -> Ss
#pragma unroll
  for (int nt = 0; nt < 8; ++nt) {
    const int n0 = (ch * 8 + nt) * 16 + lx;
    const float bias = b1[n0];
#pragma unroll
    for (int r = 0; r < 8; ++r) {
      const int row = m0 + r + half * 8;
      const float v = acc[nt][r] + bias;
      Hs[row * 260 + n0] = v > 0.f ? v : 0.f;
    }
  }
  if (ch == 0) {
    const float bias = bp[lx];
#pragma unroll
    for (int r = 0; r < 8; ++r) {
      const int row = m0 + r + half * 8;
      const float tv = t[row0 + row] + 1e-6f;
      Ss[row * 17 + lx] = -(xiacc[r] + bias) / tv;
    }
  }

  // ------------------------------------------------------------------
  // Phase 2: gflat = H @ W2 + b2  (K=256, 4 chunks, W2 L2-resident)
  // ------------------------------------------------------------------
  v8f acc2[8];
#pragma unroll
  for (int i = 0; i < 8; ++i) acc2[i] = vzero;

  issue_W2(0);
  for (int c2 = 0; c2 < NCH2; ++c2) {
    wait_async_le<0>();
    __syncthreads();
    for (int k = 0; k < KC2; k += 4) {
      const int kk = c2 * KC2 + k;
      v2f a;
      a.x = Hs[(m0 + lx) * 260 + kk + half * 2];
      a.y = Hs[(m0 + lx) * 260 + kk + half * 2 + 1];
#pragma unroll
      for (int nt = 0; nt < 8; ++nt) {
        const int n0 = (ch * 8 + nt) * 16;
        v2f b;
        b.x = W2s[(k + half * 2) * 256 + n0 + lx];
        b.y = W2s[(k + half * 2 + 1) * 256 + n0 + lx];
        acc2[nt] = WMMA4(a, b, acc2[nt]);
      }
    }
    __syncthreads();
    if (c2 + 1 < NCH2) issue_W2(c2 + 1);
  }

  // write gflat (per-row 16x16 A matrices, row-major) over Hs (same slots)
#pragma unroll
  for (int nt = 0; nt < 8; ++nt) {
    const int n0 = (ch * 8 + nt) * 16 + lx;
    const float bias = b2[n0];
#pragma unroll
    for (int r = 0; r < 8; ++r) {
      const int row = m0 + r + half * 8;
      Hs[row * 260 + n0] = acc2[nt][r] + bias;
    }
  }
  __syncthreads();

  // ------------------------------------------------------------------
  // Phase 3: g = A A^T + 0.1 I per batch row, via WMMA (same frag feeds A & B)
  // wave handles rows wave*8 .. wave*8+7
  // ------------------------------------------------------------------
  for (int rr = 0; rr < 8; ++rr) {
    const int brow = wave * 8 + rr;
    const float* Ab = Hs + brow * 260;  // 16x16 row-major
    v8f g = vzero;
#pragma unroll
    for (int kc = 0; kc < 4; ++kc) {
      v2f f;
      f.x = Ab[lx * 16 + kc * 4 + half * 2];
      f.y = Ab[lx * 16 + kc * 4 + half * 2 + 1];
      g = WMMA4(f, f, g);  // A-frag and A^T B-frag have identical lane layout
    }
#pragma unroll
    for (int r = 0; r < 8; ++r) {
      const int i = r + half * 8;  // row of g; col = lx
      const float v = g[r] + ((lx == i) ? 0.1f : 0.0f);
      Mats[brow * 257 + i * 16 + lx] = v;
    }
  }
  __syncthreads();

  // ------------------------------------------------------------------
  // Phase 4: per-row SPD solve g y = s via Cholesky (one thread per matrix)
  // ------------------------------------------------------------------
  if (tid < M_TILE) {
    float* M = Mats + tid * 257;
#pragma unroll
    for (int k = 0; k < 16; ++k) {
      const float dkk = sqrtf(M[k * 16 + k]);
      const float inv = 1.0f / dkk;
      M[k * 16 + k] = dkk;
#pragma unroll
      for (int i2 = k + 1; i2 < 16; ++i2) M[i2 * 16 + k] *= inv;
#pragma unroll
      for (int j = k + 1; j < 16; ++j) {
        const float ljk = M[j * 16 + k];
#pragma unroll
        for (int i2 = j; i2 < 16; ++i2) M[i2 * 16 + j] -= M[i2 * 16 + k] * ljk;
      }
    }
    float* s = Ss + tid * 17;
    float z[16];
#pragma unroll
    for (int i2 = 0; i2 < 16; ++i2) {  // forward: L z = s
      float v = s[i2];
#pragma unroll
      for (int j = 0; j < i2; ++j) v -= M[i2 * 16 + j] * z[j];
      z[i2] = v / M[i2 * 16 + i2];
    }
#pragma unroll
    for (int i2 = 15; i2 >= 0; --i2) {  // backward: L^T y = z
      float v = z[i2];
#pragma unroll
      for (int j = i2 + 1; j < 16; ++j) v -= M[j * 16 + i2] * z[j];
      z[i2] = v / M[i2 * 16 + i2];
      s[i2] = z[i2];
    }
  }
  __syncthreads();

  // ------------------------------------------------------------------
  // Phase 5: out = Y @ Wl + bl  (K=16, N=784 = 49 tiles of 16)
  // ch==0 waves take tiles 0..24, ch==1 take 25..48 (wave-uniform bounds)
  // ------------------------------------------------------------------
  const int ntiles = (ch == 0) ? 25 : 24;
  const int ntbase = (ch == 0) ? 0 : 25;
  for (int tix = 0; tix < ntiles; ++tix) {
    const int n0 = (ntbase + tix) * 16;
    v8f o = vzero;
#pragma unroll
    for (int kc = 0; kc < 4; ++kc) {
      const int kr = kc * 4 + half * 2;
      v2f a, b;
      a.x = Ss[(m0 + lx) * 17 + kr];
      a.y = Ss[(m0 + lx) * 17 + kr + 1];
      b.x = Wl[(size_t)kr * DDIM + n0 + lx];
      b.y = Wl[(size_t)(kr + 1) * DDIM + n0 + lx];
      o = WMMA4(a, b, o);
    }
    const float bias = bl[n0 + lx];
#pragma unroll
    for (int r = 0; r < 8; ++r) {
      const int row = row0 + m0 + r + half * 8;
      out[(size_t)row * DDIM + n0 + lx] = o[r] + bias;
    }
  }
}

extern "C" void kernel_launch(void* const* d_in, const int* in_sizes, int n_in,
                              void* d_out, int out_size, void* d_ws, size_t ws_size,
                              hipStream_t stream) {
  (void)in_sizes; (void)n_in; (void)out_size; (void)d_ws; (void)ws_size;
  const float* x  = (const float*)d_in[0];
  const float* t  = (const float*)d_in[1];
  const float* W1 = (const float*)d_in[2];
  const float* b1 = (const float*)d_in[3];
  const float* W2 = (const float*)d_in[4];
  const float* b2 = (const float*)d_in[5];
  const float* Wp = (const float*)d_in[6];
  const float* bp = (const float*)d_in[7];
  const float* Wl = (const float*)d_in[8];
  const float* bl = (const float*)d_in[9];
  float* out = (float*)d_out;

  dim3 grid(BATCH / M_TILE);  // 2048 workgroups
  dim3 block(256);            // 8 waves (wave32)
  hipLaunchKernelGGL(riemannian_diffusion_fused, grid, block, 0, stream,
                     x, t, W1, b1, W2, b2, Wp, bp, Wl, bl, out);
}